// AnchorHeadBase_635655160691
// MI455X (gfx1250) — compile-verified
//
#include <hip/hip_runtime.h>
#include <hip/hip_bf16.h>

typedef __attribute__((ext_vector_type(2))) float v2f;
typedef __attribute__((ext_vector_type(8))) float v8f;

#define BATCH      4
#define C_IN       512
#define HW         65536          // 256*256
#define A_PER      6
#define N_ANCH     (HW * A_PER)   // 393216
#define NCLS       4
#define BOXD       7
#define NHEAD      66             // 24 cls + 42 reg channels
#define PIX_BLK    32             // pixels per block
#define LDS_STRIDE 34             // 32 cols + pad so +8-row lane half hits disjoint banks

__global__ __launch_bounds__(160)
void anchor_head_fused(const float* __restrict__ x,
                       const float* __restrict__ anchors,
                       const float* __restrict__ Wc,
                       const float* __restrict__ bc,
                       const float* __restrict__ Wr,
                       const float* __restrict__ br,
                       float* __restrict__ out)
{
    __shared__ float lds[80 * LDS_STRIDE];

    const int tid  = threadIdx.x;
    const int wave = tid >> 5;            // 0..4 -> output-channel tile
    const int lane = tid & 31;
    const int bid  = blockIdx.x;
    const int b    = bid >> 11;           // 2048 blocks per batch image
    const int m0   = (bid & 2047) * PIX_BLK;

    // ---------------- GEMM: heads[o][p] = sum_c W[o][c] * x[b][c][m0+p] ----------------
    const int o0  = wave * 16;
    const int row = o0 + (lane & 15);     // output-channel row this lane supplies for A
    // rows 0..23 -> Wc, 24..65 -> Wr, 66..79 -> clamp (results never read)
    const float* wrow = (row < 24)
        ? (Wc + (size_t)row * C_IN)
        : (Wr + (size_t)(((row < NHEAD) ? row : 24) - 24) * C_IN);
    const int khalf = (lane >> 4) * 2;    // lanes 16-31 hold K+2/K+3

    const float* xcol0 = x + (size_t)b * C_IN * HW + m0 + (lane & 15);
    const float* xcol1 = xcol0 + 16;

    v8f acc0 = {};
    v8f acc1 = {};

    #pragma unroll 4
    for (int it = 0; it < C_IN / 4; ++it) {
        const int c = it * 4 + khalf;
        // A frag: 16x4 f32, lane holds W[row][c], W[row][c+1] (8B aligned)
        v2f a = *(const v2f*)(wrow + c);
        // B frags: 4x16 f32, lane holds x[c][n], x[c+1][n] for its pixel column
        const float* p0 = xcol0 + (size_t)c * HW;
        const float* p1 = xcol1 + (size_t)c * HW;
        v2f b0; b0.x = p0[0]; b0.y = p0[HW];
        v2f b1; b1.x = p1[0]; b1.y = p1[HW];
        __builtin_prefetch(p0 + (size_t)8 * HW, 0, 1);   // global_prefetch_b8 down the channel stream
        acc0 = __builtin_amdgcn_wmma_f32_16x16x4_f32(false, a, false, b0, (short)0, acc0, false, false);
        acc1 = __builtin_amdgcn_wmma_f32_16x16x4_f32(false, a, false, b1, (short)0, acc1, false, false);
    }

    // D layout: VGPR r -> M = r + 8*(lane>>4), N = lane&15
    {
        const int mrow = 8 * (lane >> 4);
        const int ncol = lane & 15;
        #pragma unroll
        for (int r = 0; r < 8; ++r) {
            lds[(o0 + mrow + r) * LDS_STRIDE + ncol]      = acc0[r];
            lds[(o0 + mrow + r) * LDS_STRIDE + 16 + ncol] = acc1[r];
        }
    }
    __syncthreads();

    // ---------------- fused epilogue: bias + permute + argmax + decode ----------------
    // anchor n = kk*65536 + m reads head channel o = c*6+kk (cls) / 24 + d*6+kk (reg) at pixel m
    const int tot = PIX_BLK * A_PER;      // 192 anchors per block
    for (int ai = tid; ai < tot; ai += 160) {
        const int p  = ai & 31;
        const int kk = ai >> 5;           // 0..5
        const int n  = kk * HW + (m0 + p);

        // argmax of logits (== argmax of softmax), first-max-wins like jnp.argmax
        int   best_c = 0;
        float best   = lds[kk * LDS_STRIDE + p] + bc[kk];
        #pragma unroll
        for (int c = 1; c < NCLS; ++c) {
            float v = lds[(c * A_PER + kk) * LDS_STRIDE + p] + bc[c * A_PER + kk];
            if (v > best) { best = v; best_c = c; }
        }
        const bool pos = best_c > 0;

        float t[BOXD];
        #pragma unroll
        for (int d = 0; d < BOXD; ++d)
            t[d] = lds[(24 + d * A_PER + kk) * LDS_STRIDE + p] + br[d * A_PER + kk];

        const float* an = anchors + (size_t)n * 7;
        const float xa = an[0], ya = an[1], za = an[2];
        const float wa = an[3], la = an[4], ha = an[5], ra = an[6];
        const float diag = sqrtf(wa * wa + la * la);

        float box[BOXD];
        box[0] = t[0] * diag + xa;
        box[1] = t[1] * diag + ya;
        box[2] = t[2] * ha + za;
        box[3] = __expf(t[3]) * wa;
        box[4] = __expf(t[4]) * la;
        box[5] = __expf(t[5]) * ha;
        box[6] = t[6] + ra;

        float* ob = out + ((size_t)b * N_ANCH + n) * 7;
        #pragma unroll
        for (int d = 0; d < BOXD; ++d) ob[d] = pos ? box[d] : 0.0f;

        out[(size_t)BATCH * N_ANCH * 7 + (size_t)b * N_ANCH + n] = (float)best_c;
        out[(size_t)BATCH * N_ANCH * 8 + (size_t)b * N_ANCH + n] = pos ? 1.0f : 0.0f;
    }
}

extern "C" void kernel_launch(void* const* d_in, const int* in_sizes, int n_in,
                              void* d_out, int out_size, void* d_ws, size_t ws_size,
                              hipStream_t stream) {
    const float* x       = (const float*)d_in[0];
    const float* anchors = (const float*)d_in[1];
    const float* Wc      = (const float*)d_in[2];
    const float* bc      = (const float*)d_in[3];
    const float* Wr      = (const float*)d_in[4];
    const float* br      = (const float*)d_in[5];
    float* out = (float*)d_out;

    const int blocks = BATCH * (HW / PIX_BLK);   // 8192
    anchor_head_fused<<<blocks, 160, 0, stream>>>(x, anchors, Wc, bc, Wr, br, out);
}